// EMAVectorQuantizer_80229989089576
// MI455X (gfx1250) — compile-verified
//
#include <hip/hip_runtime.h>
#include <hip/hip_bf16.h>

typedef __attribute__((ext_vector_type(2))) float v2f;
typedef __attribute__((ext_vector_type(8))) float v8f;
typedef __attribute__((address_space(3))) float lds_f;

#define NUM_EMB   1024
#define EMB_DIM   256
#define N_ROWS    65536            // 64*1024
#define DECAY_F   0.99f
#define ONE_M_DECAY 0.01f
#define EPS_F     1e-05f
#define TOT_ELEMS 16777216.0f      // N_ROWS * EMB_DIM

#define B_PAD     4                // LDS row padding (floats) -> 260-float rows, bank-conflict free
#define B_ROW     (EMB_DIM + B_PAD)

// ---------------- workspace layout (in floats) ----------------
#define WS_COUNTS 0
#define WS_DW     1024
#define WS_LOSS   263168
#define WS_NSUM   263169
#define WS_EHALF  263170
#define WS_NEWCS  264194
#define WS_IDX    265218
#define WS_ZERO_N 263170           // zero counts+dw+loss+nsum

// ---------------- async global->LDS helpers ----------------
__device__ __forceinline__ void async_b128(unsigned lds_off, const void* gaddr) {
    asm volatile("global_load_async_to_lds_b128 %0, %1, off"
                 :: "v"(lds_off), "v"((unsigned long long)(size_t)gaddr)
                 : "memory");
}
__device__ __forceinline__ void wait_async0() {
#if __has_builtin(__builtin_amdgcn_s_wait_asynccnt)
    __builtin_amdgcn_s_wait_asynccnt(0);
#else
    asm volatile("s_wait_asynccnt 0x0" ::: "memory");
#endif
}

// ---------------- kernel 0: zero accumulators ----------------
__global__ void vq_zero_kernel(float* ws) {
    int i = blockIdx.x * 256 + threadIdx.x;
    if (i < WS_ZERO_N) ws[i] = 0.0f;
}

// ---------------- kernel 1: 0.5*||e_k||^2 ----------------
__global__ __launch_bounds__(256) void vq_enorm_kernel(const float* __restrict__ e,
                                                       float* __restrict__ ehalf) {
    int k = blockIdx.x, d = threadIdx.x;
    float v = e[(size_t)k * EMB_DIM + d];
    __shared__ float red[256];
    red[d] = v * v;
    __syncthreads();
    for (int s = 128; s > 0; s >>= 1) {
        if (d < s) red[d] += red[d + s];
        __syncthreads();
    }
    if (d == 0) ehalf[k] = 0.5f * red[0];
}

// ---------------- kernel 2: fp32 WMMA scores + argmin ----------------
// 8 waves/block, 16 rows/wave (128 rows/block). B tiles (16 codes x 256 dims)
// double-buffered in LDS via async global->LDS copies shared by all 8 waves.
__global__ __launch_bounds__(256) void vq_argmin_wmma_kernel(
        const float* __restrict__ x, const float* __restrict__ e,
        const float* __restrict__ ehalf, int* __restrict__ idx_out,
        float* __restrict__ idx_f_out) {
    __shared__ float btile[2][16 * B_ROW];     // 2 x 16.25 KB

    const int tid  = threadIdx.x;
    const int wave = tid >> 5;
    const int lane = tid & 31;
    const int m  = lane & 15;   // A row / B code / C column within tile
    const int kh = lane >> 4;   // which K-pair of the 16x4 fragment this lane holds

    const int row0 = blockIdx.x * 128 + wave * 16;

    // A fragments: lane holds x[row0+m][4t + kh*2 + {0,1}] — full D in registers
    const float* arow = x + (size_t)(row0 + m) * EMB_DIM + kh * 2;
    v2f afrag[64];
#pragma unroll
    for (int t = 0; t < 64; ++t)
        afrag[t] = *(const v2f*)(arow + t * 4);

    float bestVal[8];
    int   bestIdx[8];
#pragma unroll
    for (int j = 0; j < 8; ++j) { bestVal[j] = -3.402823466e38f; bestIdx[j] = 0; }

    // async copy of B tile c into buffer buf: 16 rows x 1KB, padded LDS rows.
    // 1024 b128 transfers; 256 threads x 4 each.
    const unsigned ldsbase[2] = {
        (unsigned)(size_t)(lds_f*)&btile[0][0],
        (unsigned)(size_t)(lds_f*)&btile[1][0]
    };
    const char* ebytes = (const char*)e;

    {   // prologue: tile 0 -> buf 0
#pragma unroll
        for (int r = 0; r < 4; ++r) {
            int li = r * 256 + tid;
            int brow = li >> 6, bcol = li & 63;
            async_b128(ldsbase[0] + (unsigned)(brow * (B_ROW * 4) + bcol * 16),
                       ebytes + (size_t)brow * 1024 + bcol * 16);
        }
    }

    for (int c = 0; c < 64; ++c) {
        const int cur = c & 1;
        wait_async0();              // this wave's slice of tile c has landed in LDS
        __syncthreads();            // all waves' slices visible; buf[1-cur] now free

        if (c + 1 < 64) {           // overlap: prefetch tile c+1 into the other buffer
            const char* src = ebytes + (size_t)(c + 1) * 16384;
#pragma unroll
            for (int r = 0; r < 4; ++r) {
                int li = r * 256 + tid;
                int brow = li >> 6, bcol = li & 63;
                async_b128(ldsbase[1 - cur] + (unsigned)(brow * (B_ROW * 4) + bcol * 16),
                           src + (size_t)brow * 1024 + bcol * 16);
            }
        }

        // 16x16 score tile over D=256: 64 chained WMMAs, two chains to break RAW
        const float* bsm = &btile[cur][m * B_ROW + kh * 2];
        v8f acc0 = {}; v8f acc1 = {};
#pragma unroll
        for (int t = 0; t < 64; t += 2) {
            v2f b0 = *(const v2f*)(bsm + t * 4);
            v2f b1 = *(const v2f*)(bsm + (t + 1) * 4);
            acc0 = __builtin_amdgcn_wmma_f32_16x16x4_f32(
                false, afrag[t],     false, b0, (short)0, acc0, false, false);
            acc1 = __builtin_amdgcn_wmma_f32_16x16x4_f32(
                false, afrag[t + 1], false, b1, (short)0, acc1, false, false);
        }

        // score = x.e - 0.5||e||^2 ; argmin(dist) == argmax(score)
        const float hn  = ehalf[c * 16 + m];
        const int   col = c * 16 + m;
#pragma unroll
        for (int j = 0; j < 8; ++j) {
            float s = (acc0[j] + acc1[j]) - hn;
            if (s > bestVal[j]) { bestVal[j] = s; bestIdx[j] = col; }
        }
    }

    // Butterfly argmax across the 16 lanes of each half (N is lane-striped).
#pragma unroll
    for (int msk = 8; msk >= 1; msk >>= 1) {
#pragma unroll
        for (int j = 0; j < 8; ++j) {
            float ov = __shfl_xor(bestVal[j], msk, 32);
            int   oi = __shfl_xor(bestIdx[j], msk, 32);
            if (ov > bestVal[j] || (ov == bestVal[j] && oi < bestIdx[j])) {
                bestVal[j] = ov; bestIdx[j] = oi;
            }
        }
    }
    // lanes 0..15 hold rows 0..7 (C VGPR j -> row j); lanes 16..31 hold rows 8..15
    if (m == 0) {
        int base = row0 + kh * 8;
#pragma unroll
        for (int j = 0; j < 8; ++j) {
            idx_out[base + j]   = bestIdx[j];
            idx_f_out[base + j] = (float)bestIdx[j];
        }
    }
}

// ---------------- kernel 3: gather + loss + segment sums ----------------
__global__ __launch_bounds__(256) void vq_gather_stats_kernel(
        const float* __restrict__ x, const float* __restrict__ e,
        const int* __restrict__ idx, float* __restrict__ qout,
        float* __restrict__ dw, float* __restrict__ counts,
        float* __restrict__ loss_acc) {
    int row = blockIdx.x, d = threadIdx.x;
    int k = idx[row];
    float xv = x[(size_t)row * EMB_DIM + d];
    float ev = e[(size_t)k * EMB_DIM + d];
    qout[(size_t)row * EMB_DIM + d] = ev;   // quantized_ste == quantized (value-wise)
    float diff = ev - xv;
    __shared__ float red[256];
    red[d] = diff * diff;
    __syncthreads();
    for (int s = 128; s > 0; s >>= 1) {
        if (d < s) red[d] += red[d + s];
        __syncthreads();
    }
    if (d == 0) {
        atomicAdd(loss_acc, red[0]);
        atomicAdd(&counts[k], 1.0f);
    }
    atomicAdd(&dw[(size_t)k * EMB_DIM + d], xv);
}

// ---------------- kernel 4: new_cs (pre-norm) + n = sum(new_cs) ----------------
__global__ __launch_bounds__(256) void vq_newcs_kernel(
        const float* __restrict__ ema_cs, const float* __restrict__ counts,
        float* __restrict__ newcs, float* __restrict__ nsum) {
    int k = blockIdx.x * 256 + threadIdx.x;
    float v = ema_cs[k] * DECAY_F + ONE_M_DECAY * counts[k];
    newcs[k] = v;
    __shared__ float red[256];
    red[threadIdx.x] = v;
    __syncthreads();
    for (int s = 128; s > 0; s >>= 1) {
        if (threadIdx.x < s) red[threadIdx.x] += red[threadIdx.x + s];
        __syncthreads();
    }
    if (threadIdx.x == 0) atomicAdd(nsum, red[0]);
}

// ---------------- kernel 5: finalize outputs ----------------
__global__ __launch_bounds__(256) void vq_finalize_kernel(
        const float* __restrict__ ema_w, const float* __restrict__ dw,
        const float* __restrict__ newcs, const float* __restrict__ nsum_p,
        const float* __restrict__ loss_p, float* __restrict__ out_loss,
        float* __restrict__ out_newcs, float* __restrict__ out_emaw,
        float* __restrict__ out_upd) {
    int i = blockIdx.x * 256 + threadIdx.x;    // 0 .. K*D-1
    int k = i >> 8;
    float nsum = *nsum_p;
    float denom = nsum + (float)NUM_EMB * EPS_F;
    float ncs = (newcs[k] + EPS_F) / denom * nsum;
    float nw = ema_w[i] * DECAY_F + ONE_M_DECAY * dw[i];
    out_emaw[i] = nw;
    out_upd[i]  = nw / ncs;
    if (i < NUM_EMB) out_newcs[i] = (newcs[i] + EPS_F) / denom * nsum;
    if (i == 0)      out_loss[0]  = 1.25f * loss_p[0] / TOT_ELEMS;
}

extern "C" void kernel_launch(void* const* d_in, const int* in_sizes, int n_in,
                              void* d_out, int out_size, void* d_ws, size_t ws_size,
                              hipStream_t stream) {
    const float* x      = (const float*)d_in[0];   // [65536, 256]
    const float* e      = (const float*)d_in[1];   // [1024, 256]
    const float* ema_cs = (const float*)d_in[2];   // [1024]
    const float* ema_w  = (const float*)d_in[3];   // [1024, 256]

    float* out = (float*)d_out;
    float* out_q     = out;                        // 16777216
    float* out_loss  = out + 16777216;             // 1
    float* out_idx   = out + 16777217;             // 65536 (as float)
    float* out_newcs = out + 16842753;             // 1024
    float* out_emaw  = out + 16843777;             // 262144
    float* out_upd   = out + 17105921;             // 262144

    float* ws      = (float*)d_ws;
    float* counts  = ws + WS_COUNTS;
    float* dw      = ws + WS_DW;
    float* loss    = ws + WS_LOSS;
    float* nsum    = ws + WS_NSUM;
    float* ehalf   = ws + WS_EHALF;
    float* newcs   = ws + WS_NEWCS;
    int*   idx     = (int*)(ws + WS_IDX);

    vq_zero_kernel<<<(WS_ZERO_N + 255) / 256, 256, 0, stream>>>(ws);
    vq_enorm_kernel<<<NUM_EMB, 256, 0, stream>>>(e, ehalf);
    vq_argmin_wmma_kernel<<<N_ROWS / 128, 256, 0, stream>>>(x, e, ehalf, idx, out_idx);
    vq_gather_stats_kernel<<<N_ROWS, 256, 0, stream>>>(x, e, idx, out_q, dw, counts, loss);
    vq_newcs_kernel<<<NUM_EMB / 256, 256, 0, stream>>>(ema_cs, counts, newcs, nsum);
    vq_finalize_kernel<<<(NUM_EMB * EMB_DIM) / 256, 256, 0, stream>>>(
        ema_w, dw, newcs, nsum, loss, out_loss, out_newcs, out_emaw, out_upd);
}